// ModulatedDeformConvWithOff_40441412059322
// MI455X (gfx1250) — compile-verified
//
#include <hip/hip_runtime.h>
#include <hip/hip_bf16.h>
#include <math.h>

// Problem constants (from reference setup_inputs)
#define BN    4
#define CIN   64
#define HH    160
#define WW    160
#define COUT  64
#define KKT   9            // K*K taps
#define OFFC  27           // 3*K*K offset/mask channels
#define KDIM  576          // CIN * KKT  (GEMM K dimension)
#define HW    (HH*WW)

typedef __attribute__((ext_vector_type(2))) float v2f;
typedef __attribute__((ext_vector_type(8))) float v8f;

// ---------------------------------------------------------------------------
// Kernel 1: 3x3 conv producing 27 offset/mask channels -> om[B][27][H][W]
// Weights (27*576 floats = 62KB) staged in LDS; each thread computes one pixel
// and keeps 27 fp32 accumulators in VGPRs.
// ---------------------------------------------------------------------------
__global__ __launch_bounds__(256) void dcn_offset_conv(
    const float* __restrict__ x,      // [B][CIN][H][W]
    const float* __restrict__ w_off,  // [27][CIN][3][3]
    const float* __restrict__ b_off,  // [27]
    float* __restrict__ om)           // [B][27][H][W]
{
  __shared__ float wsm[OFFC * KDIM];  // 62,208 bytes
  for (int i = threadIdx.x; i < OFFC * KDIM; i += 256) wsm[i] = w_off[i];
  __syncthreads();

  const int pix = blockIdx.x * 256 + threadIdx.x;  // over B*H*W (exact multiple)
  if (pix >= BN * HW) return;
  const int b  = pix / HW;
  const int hw = pix - b * HW;
  const int oh = hw / WW;
  const int ow = hw - oh * WW;

  float acc[OFFC];
#pragma unroll
  for (int o = 0; o < OFFC; ++o) acc[o] = b_off[o];

  const float* xb = x + (size_t)b * CIN * HW;
  for (int c = 0; c < CIN; ++c) {
    const float* xc = xb + c * HW;
#pragma unroll
    for (int ky = 0; ky < 3; ++ky) {
      const int iy = oh + ky - 1;
      if (iy < 0 || iy >= HH) continue;
#pragma unroll
      for (int kx = 0; kx < 3; ++kx) {
        const int ix = ow + kx - 1;
        if (ix < 0 || ix >= WW) continue;
        const float v  = xc[iy * WW + ix];
        const int   wi = c * KKT + ky * 3 + kx;
#pragma unroll
        for (int o = 0; o < OFFC; ++o)
          acc[o] = fmaf(v, wsm[o * KDIM + wi], acc[o]);
      }
    }
  }

  float* omp = om + ((size_t)b * OFFC) * HW + hw;
#pragma unroll
  for (int o = 0; o < OFFC; ++o) omp[(size_t)o * HW] = acc[o];
}

// ---------------------------------------------------------------------------
// Kernel 2: fused bilinear sampling + WMMA fp32 GEMM.
// One block (128 thr = 4 wave32) handles 16 consecutive output pixels of one
// row. Phase 1 builds val[576][16] in LDS; phase 2: each wave computes a
// 16-channel tile of out = W[64x576] * val[576x16] via V_WMMA_F32_16X16X4_F32.
// ---------------------------------------------------------------------------
__global__ __launch_bounds__(128) void dcn_sample_gemm(
    const float* __restrict__ x,      // [B][CIN][H][W]
    const float* __restrict__ om,     // [B][27][H][W]
    const float* __restrict__ w_dcn,  // [COUT][CIN][3][3] == [COUT][576]
    const float* __restrict__ b_dcn,  // [COUT]
    float* __restrict__ out)          // [B][COUT][H][W]
{
  __shared__ float valsm[KDIM * 16];   // 36,864 bytes: val[c*9+k][pixel]
  __shared__ int   sy0[KKT * 16];
  __shared__ int   sx0[KKT * 16];
  __shared__ float sdy[KKT * 16];
  __shared__ float sdx[KKT * 16];
  __shared__ float smk[KKT * 16];

  // tile decode: blockIdx.x over B * H * (W/16)
  const int tilesW = WW / 16;
  int t  = blockIdx.x;
  const int b  = t / (HH * tilesW);   t -= b * (HH * tilesW);
  const int oh = t / tilesW;
  const int w0 = (t - oh * tilesW) * 16;

  const int tid = threadIdx.x;

  // ---- Phase 1a: per (tap k, pixel p) geometry + mask ---------------------
  for (int pid = tid; pid < KKT * 16; pid += 128) {
    const int k  = pid >> 4;
    const int p  = pid & 15;
    const int ow = w0 + p;
    const int ky = k / 3;
    const int kx = k - ky * 3;
    const size_t obase = ((size_t)b * OFFC) * HW + (size_t)oh * WW + ow;
    const float oy = om[obase + (size_t)(2 * k)     * HW];
    const float ox = om[obase + (size_t)(2 * k + 1) * HW];
    const float mz = om[obase + (size_t)(18 + k)    * HW];
    const float m  = 1.0f / (1.0f + __expf(-mz));
    const float yf = (float)(oh - 1 + ky) + oy;
    const float xf = (float)(ow - 1 + kx) + ox;
    const float y0 = floorf(yf);
    const float x0 = floorf(xf);
    sy0[pid] = (int)y0;
    sx0[pid] = (int)x0;
    sdy[pid] = yf - y0;
    sdx[pid] = xf - x0;
    smk[pid] = m;
  }
  __syncthreads();

  // ---- Phase 1b: bilinear gather for all (pair, channel) tasks ------------
  for (int task = tid; task < KKT * 16 * CIN; task += 128) {
    const int pid = task % (KKT * 16);
    const int c   = task / (KKT * 16);
    const int k   = pid >> 4;
    const int p   = pid & 15;
    const int y0  = sy0[pid];
    const int x0  = sx0[pid];
    const float dy = sdy[pid];
    const float dx = sdx[pid];
    const float m  = smk[pid];
    const float* xc = x + ((size_t)b * CIN + c) * HW;

    const int y1 = y0 + 1, x1 = x0 + 1;
    const bool vy0 = (y0 >= 0) & (y0 < HH);
    const bool vy1 = (y1 >= 0) & (y1 < HH);
    const bool vx0 = (x0 >= 0) & (x0 < WW);
    const bool vx1 = (x1 >= 0) & (x1 < WW);
    const float g00 = (vy0 & vx0) ? xc[y0 * WW + x0] : 0.0f;
    const float g01 = (vy0 & vx1) ? xc[y0 * WW + x1] : 0.0f;
    const float g10 = (vy1 & vx0) ? xc[y1 * WW + x0] : 0.0f;
    const float g11 = (vy1 & vx1) ? xc[y1 * WW + x1] : 0.0f;

    float v = g00 * (1.0f - dy) * (1.0f - dx)
            + g01 * (1.0f - dy) * dx
            + g10 * dy * (1.0f - dx)
            + g11 * dy * dx;
    valsm[(c * KKT + k) * 16 + p] = v * m;
  }
  __syncthreads();

  // ---- Phase 2: out[64][16] = W[64][576] x val[576][16] via fp32 WMMA -----
  // All 128 lanes active => EXEC all ones (WMMA requirement).
  const int wave = tid >> 5;          // M-tile: output channels [wave*16, +16)
  const int lane = tid & 31;
  const int lm   = lane & 15;
  const int hi   = lane >> 4;         // 0: K0/K1 half, 1: K2/K3 half

  const float* wrow = w_dcn + (size_t)(wave * 16 + lm) * KDIM;

  v8f acc = {};
  for (int kb = 0; kb < KDIM / 4; ++kb) {
    const int kbase = kb * 4 + hi * 2;
    // A: 16x4 fp32 (lanes 0-15: M rows K0/K1; lanes 16-31: K2/K3)
    v2f a; a.x = wrow[kbase]; a.y = wrow[kbase + 1];
    // B: 4x16 fp32 (lanes 0-15: N cols K0/K1; lanes 16-31: K2/K3)
    v2f bm; bm.x = valsm[(kbase + 0) * 16 + lm];
            bm.y = valsm[(kbase + 1) * 16 + lm];
    acc = __builtin_amdgcn_wmma_f32_16x16x4_f32(
        /*neg_a=*/false, a, /*neg_b=*/false, bm,
        /*c_mod=*/(short)0, acc, /*reuse_a=*/false, /*reuse_b=*/false);
  }

  // D layout: VGPR i, lanes 0-15 -> M=i ; lanes 16-31 -> M=i+8 ; N = lane&15
  const int ow = w0 + lm;
  float* outp = out + ((size_t)b * COUT) * HW + (size_t)oh * WW + ow;
#pragma unroll
  for (int i = 0; i < 8; ++i) {
    const int ch = wave * 16 + hi * 8 + i;
    outp[(size_t)ch * HW] = acc[i] + b_dcn[ch];
  }
}

// ---------------------------------------------------------------------------
extern "C" void kernel_launch(void* const* d_in, const int* in_sizes, int n_in,
                              void* d_out, int out_size, void* d_ws, size_t ws_size,
                              hipStream_t stream) {
  const float* x     = (const float*)d_in[0];  // [4][64][160][160]
  const float* w_off = (const float*)d_in[1];  // [27][64][3][3]
  const float* b_off = (const float*)d_in[2];  // [27]
  const float* w_dcn = (const float*)d_in[3];  // [64][64][3][3]
  const float* b_dcn = (const float*)d_in[4];  // [64]
  float* out = (float*)d_out;                  // [4][64][160][160]
  float* om  = (float*)d_ws;                   // [4][27][160][160] = 11.06 MB

  // Kernel 1: offset conv -> om
  const int pixTotal = BN * HW;                     // 102400
  dcn_offset_conv<<<(pixTotal + 255) / 256, 256, 0, stream>>>(x, w_off, b_off, om);

  // Kernel 2: fused bilinear sampling + WMMA GEMM
  const int nTiles = BN * HH * (WW / 16);           // 6400
  dcn_sample_gemm<<<nTiles, 128, 0, stream>>>(x, om, w_dcn, b_dcn, out);
}